// NaiveAssemble_56564719288570
// MI455X (gfx1250) — compile-verified
//
#include <hip/hip_runtime.h>
#include <stdint.h>
#include <math.h>

#if __has_include(<hip/amd_detail/amd_gfx1250_TDM.h>)
#define TDM_SIX_ARG 1
#endif

#if __has_builtin(__builtin_amdgcn_tensor_load_to_lds) && __has_builtin(__builtin_amdgcn_s_wait_tensorcnt)
#define HAVE_TDM 1
#endif

typedef float v2f __attribute__((ext_vector_type(2)));
typedef float v8f __attribute__((ext_vector_type(8)));
typedef unsigned int v4u __attribute__((ext_vector_type(4)));
typedef int v4i __attribute__((ext_vector_type(4)));
typedef int v8i __attribute__((ext_vector_type(8)));

#define KSEL 16
#define NPIX 4096
#define NPIX_LOG2 12
#define CCH 256
#define TPB 256
#define EPT 16             /* elements per thread (512-elem wave segment / 32 lanes) */
#define NWAVE (TPB / 32)   /* 8 waves */
#define SEGLEN 512         /* NPIX / NWAVE */

// ---------------------------------------------------------------------------
// feat [B, C, N] -> featT [B, N, C]  (classic LDS tiled transpose)
// ---------------------------------------------------------------------------
__global__ __launch_bounds__(256) void feat_transpose_kernel(
    const float* __restrict__ feat, float* __restrict__ featT) {
  __shared__ float tile[32][33];
  const int b = blockIdx.z;
  const int j0 = blockIdx.x * 32;
  const int c0 = blockIdx.y * 32;
  const int tx = threadIdx.x;       // 0..31
  const int ty = threadIdx.y;       // 0..7
  const float* fb = feat + (size_t)b * CCH * NPIX;
  float* ftb = featT + (size_t)b * NPIX * CCH;
#pragma unroll
  for (int i = 0; i < 4; ++i) {
    const int c = c0 + ty + 8 * i;
    tile[ty + 8 * i][tx] = fb[(size_t)c * NPIX + (j0 + tx)];
  }
  __syncthreads();
#pragma unroll
  for (int i = 0; i < 4; ++i) {
    const int j = j0 + ty + 8 * i;
    ftb[(size_t)j * CCH + (c0 + tx)] = tile[tx][ty + 8 * i];
  }
}

// ---------------------------------------------------------------------------
// One workgroup per (b, n) row:
//   TDM row -> LDS; per-wave top-16 (barrier-free shfl butterflies);
//   wave-0 merge of 128 candidates + softmax; WMMA f32 assemble.
// ---------------------------------------------------------------------------
__global__ __launch_bounds__(TPB) void topk_assemble_kernel(
    const float* __restrict__ aff,   // [B, N, N]
    const float* __restrict__ fsrc,  // featT [B,N,C] if transposed else feat [B,C,N]
    float* __restrict__ out,         // [B, C, N]
    int transposed) {
  __shared__ float candv[NWAVE * KSEL];   // per-wave top-16 values
  __shared__ int   candj[NWAVE * KSEL];   // per-wave top-16 indices
  __shared__ int   resj[KSEL];            // global top-16 indices (descending)
  __shared__ float wgt[KSEL];             // softmax weights

  const int tid  = threadIdx.x;
  const int lane = tid & 31;
  const int wv   = tid >> 5;
  const int row  = blockIdx.x;            // 0 .. B*N-1
  const int b    = row >> NPIX_LOG2;
  const int n    = row & (NPIX - 1);
  const float* rowg = aff + (size_t)row * NPIX;
  const int seg = wv * SEGLEN;            // this wave's 512-element segment

  float vals[EPT];

#if HAVE_TDM
  __shared__ float rowbuf[NPIX];          // 16 KB row staging
  if (tid < 32) {                         // wave 0: one TDM issue per WG (EXEC-independent)
    const uint64_t ga = (uint64_t)(uintptr_t)rowg;
    const uint32_t lds_off = (uint32_t)(uintptr_t)(&rowbuf[0]); // LDS aperture: low 32b = offset
    v4u g0;
    g0[0] = 1u;                                                // count=1, user descriptor
    g0[1] = lds_off;                                           // lds_addr
    g0[2] = (uint32_t)ga;                                      // global_addr[31:0]
    g0[3] = ((uint32_t)(ga >> 32) & 0x01FFFFFFu) | (2u << 30); // addr[56:32], type=2
    v8i g1;
    g1[0] = (int)(2u << 16);                                   // data_size = 4B (wg_mask=0)
    g1[1] = (int)((uint32_t)NPIX << 16);                       // tensor_dim0 = 4096 (low16)
    g1[2] = (int)(1u << 16);                                   // tensor_dim1 = 1
    g1[3] = (int)((uint32_t)NPIX << 16);                       // tile_dim0 = 4096
    g1[4] = 1;                                                 // tile_dim1 = 1, tile_dim2 = 0
    g1[5] = NPIX;                                              // tensor_dim0_stride low32
    g1[6] = (int)((uint32_t)NPIX << 16);                       // dim1_stride lo16 (dim0 hi=0)
    g1[7] = 0;                                                 // dim1_stride hi
    v4i gz = {0, 0, 0, 0};
#if TDM_SIX_ARG
    v8i gz8 = {0, 0, 0, 0, 0, 0, 0, 0};
    __builtin_amdgcn_tensor_load_to_lds(g0, g1, gz, gz, gz8, 0);
#else
    __builtin_amdgcn_tensor_load_to_lds(g0, g1, gz, gz, 0);
#endif
    __builtin_amdgcn_s_wait_tensorcnt(0);
  }
  __syncthreads();
#pragma unroll
  for (int i = 0; i < EPT; ++i) vals[i] = rowbuf[seg + i * 32 + lane]; // conflict-free
#else
#pragma unroll
  for (int i = 0; i < EPT; ++i) vals[i] = rowg[seg + i * 32 + lane];   // coalesced fallback
#endif

  // ---- Phase A: per-wave top-16 of its 512-element segment (no barriers) ----
  // element index: j = seg + i*32 + lane  ->  i = (j & 511) >> 5, owner lane = j & 31
  {
    unsigned removed = 0u;
    float lv = vals[0];
    int li = 0;
#pragma unroll
    for (int i = 1; i < EPT; ++i)
      if (vals[i] > lv) { lv = vals[i]; li = i; }
    int lj = seg + li * 32 + lane;

    for (int t = 0; t < KSEL; ++t) {
      float v = lv;
      int j = lj;
#pragma unroll
      for (int off = 16; off > 0; off >>= 1) {     // butterfly: all lanes get winner
        const float ov = __shfl_xor(v, off, 32);
        const int oj = __shfl_xor(j, off, 32);
        if (ov > v || (ov == v && oj < j)) { v = ov; j = oj; }
      }
      if (lane == 0) { candv[wv * KSEL + t] = v; candj[wv * KSEL + t] = j; }
      if ((j & 31) == lane) {                      // owner removes & recomputes
        removed |= 1u << ((j & (SEGLEN - 1)) >> 5);
        lv = -__builtin_inff();
        li = 0;
#pragma unroll
        for (int i = 0; i < EPT; ++i)
          if (!((removed >> i) & 1u) && vals[i] > lv) { lv = vals[i]; li = i; }
        lj = seg + li * 32 + lane;
      }
    }
  }
  __syncthreads();

  // ---- Phase B: wave 0 merges 128 candidates -> global top-16 + softmax ----
  if (tid < 32) {
    float cv[4];
    int cj[4];
#pragma unroll
    for (int q = 0; q < 4; ++q) {
      cv[q] = candv[lane + 32 * q];
      cj[q] = candj[lane + 32 * q];
    }
    unsigned rem = 0u;
    float myv = 0.f;                       // lane t will hold t-th selected value

    for (int t = 0; t < KSEL; ++t) {
      float bv = -__builtin_inff();
      int bj = 0x7fffffff;
      int bq = 0;
#pragma unroll
      for (int q = 0; q < 4; ++q)
        if (!((rem >> q) & 1u) &&
            (cv[q] > bv || (cv[q] == bv && cj[q] < bj))) {
          bv = cv[q]; bj = cj[q]; bq = q;
        }
      int bs = (bq << 5) | lane;           // candidate slot, carried through reduce
#pragma unroll
      for (int off = 16; off > 0; off >>= 1) {
        const float ov = __shfl_xor(bv, off, 32);
        const int oj = __shfl_xor(bj, off, 32);
        const int os = __shfl_xor(bs, off, 32);
        if (ov > bv || (ov == bv && oj < bj)) { bv = ov; bj = oj; bs = os; }
      }
      if (lane == 0) resj[t] = bj;
      myv = (lane == t) ? bv : myv;
      if ((bs & 31) == lane) rem |= 1u << (bs >> 5);
    }

    // softmax over the 16 selected values (lane l holds value l; lane 0 = max)
    const float m = __shfl(myv, 0, 32);
    float e = (lane < KSEL) ? expf(myv - m) : 0.f;
    float s = e;
#pragma unroll
    for (int off = 16; off > 0; off >>= 1) s += __shfl_xor(s, off, 32);
    if (lane < KSEL) wgt[lane] = e / s;
  }
  __syncthreads();

  // ---- weighted gather-sum via V_WMMA_F32_16X16X4_F32 ----
  // A[m,k] = w[k] (replicated over M), B[k,c] = feat[idx[k], c].
  // f32 16x16x4 fragments: VGPR0 = {K0 | K2}, VGPR1 = {K1 | K3} across lane halves.
  const int hk = (lane < 16) ? 0 : 2;     // lane-half K offset within a 4-k chunk
  const int cl = lane & 15;               // N index -> channel-in-chunk
  const float* fb = fsrc + (size_t)b * NPIX * CCH;          // same for both layouts
  const size_t strj = transposed ? (size_t)CCH : (size_t)1; // j stride (uniform)

  for (int it = 0; it < 2; ++it) {
    const int c = (wv * 2 + it) * 16 + cl;                  // channel 0..255
    const size_t offc = transposed ? (size_t)c : (size_t)c * NPIX;
    v8f acc = {0.f, 0.f, 0.f, 0.f, 0.f, 0.f, 0.f, 0.f};
#pragma unroll
    for (int kc = 0; kc < 4; ++kc) {
      const int k0 = kc * 4 + hk;
      const int j0 = resj[k0];
      const int j1 = resj[k0 + 1];
      const float b0 = fb[offc + (size_t)j0 * strj];
      const float b1 = fb[offc + (size_t)j1 * strj];
      v2f av; av[0] = wgt[k0]; av[1] = wgt[k0 + 1];
      v2f bv; bv[0] = b0; bv[1] = b1;
      acc = __builtin_amdgcn_wmma_f32_16x16x4_f32(
          false, av, false, bv, (short)0, acc, false, false);
    }
    if (lane < 16) out[((size_t)b * CCH + c) * NPIX + n] = acc[0];
  }
}

// ---------------------------------------------------------------------------
extern "C" void kernel_launch(void* const* d_in, const int* in_sizes, int n_in,
                              void* d_out, int out_size, void* d_ws, size_t ws_size,
                              hipStream_t stream) {
  const float* aff = (const float*)d_in[0];   // [B, N, N] fp32
  const float* feat = (const float*)d_in[1];  // [B, C, H, W] fp32
  float* out = (float*)d_out;                 // [B, C, H, W] fp32

  const int B = in_sizes[1] / (CCH * NPIX);
  const size_t ftBytes = (size_t)B * NPIX * CCH * sizeof(float);

  int transposed = (d_ws != nullptr && ws_size >= ftBytes) ? 1 : 0;
  const float* fsrc = feat;
  if (transposed) {
    dim3 g(NPIX / 32, CCH / 32, B);
    feat_transpose_kernel<<<g, dim3(32, 8, 1), 0, stream>>>(feat, (float*)d_ws);
    fsrc = (const float*)d_ws;
  }

  topk_assemble_kernel<<<dim3(B * NPIX, 1, 1), dim3(TPB, 1, 1), 0, stream>>>(
      aff, fsrc, out, transposed);
}